// DGCNN_18365280157986
// MI455X (gfx1250) — compile-verified
//
#include <hip/hip_runtime.h>
#include <hip/hip_bf16.h>
#include <math.h>

typedef float v2f __attribute__((ext_vector_type(2)));
typedef float v8f __attribute__((ext_vector_type(8)));

#define F_IN 128
#define H1C 64
#define NCLS 18
#define KTOP 30
#define PCOLS 65
#define PLEN 1950      // KTOP*PCOLS
#define FLAT2 192
#define NGR 500
#define MPAD 512       // fc1 row padding (multiple of 16 >= NGR)

// ---------------- elementwise helpers ----------------
__global__ void k_fill_f(float* p, int n, float v) {
  int i = blockIdx.x * blockDim.x + threadIdx.x;
  if (i < n) p[i] = v;
}
__global__ void k_zero_i(int* p, int n) {
  int i = blockIdx.x * blockDim.x + threadIdx.x;
  if (i < n) p[i] = 0;
}
__global__ void k_tanh(float* p, int n) {
  int i = blockIdx.x * blockDim.x + threadIdx.x;
  if (i < n) p[i] = tanhf(p[i]);
}

// ---------------- degree / normalization ----------------
__global__ void k_deg(const int* __restrict__ row, const int* __restrict__ col,
                      float* __restrict__ deg, int E) {
  int e = blockIdx.x * blockDim.x + threadIdx.x;
  if (e >= E) return;
  int r = row[e], c = col[e];
  if (r != c) atomicAdd(&deg[c], 1.0f);
}
__global__ void k_dis(const float* __restrict__ deg, float* __restrict__ dis, int n) {
  int i = blockIdx.x * blockDim.x + threadIdx.x;
  if (i < n) dis[i] = rsqrtf(deg[i]);
}

// ---------------- WMMA f32 GEMM: xw1 = x @ W1 (N x 128 @ 128 x 64) ----------------
// blockDim = 128 (4 waves). Block handles 16 rows (N divisible by 16), wave w handles
// cols [16w,16w+16). B tile held in registers; one b64 A load per WMMA, fully unrolled.
__global__ void k_gemm1(const float* __restrict__ x, const float* __restrict__ W1,
                        float* __restrict__ xw1) {
  int wave = threadIdx.x >> 5;
  int lane = threadIdx.x & 31;
  int row0 = blockIdx.x * 16;
  int colBase = wave * 16;
  int mlane = lane & 15;
  int khalf = (lane >= 16) ? 2 : 0;

  v2f breg[32];
#pragma unroll
  for (int s = 0; s < 32; ++s) {
    int ka = 4 * s + khalf;
    breg[s].x = W1[(size_t)ka * H1C + colBase + mlane];
    breg[s].y = W1[(size_t)(ka + 1) * H1C + colBase + mlane];
  }

  const float* xrow = x + (size_t)(row0 + mlane) * F_IN + khalf;
  v8f acc = {};
#pragma unroll
  for (int s = 0; s < 32; ++s) {
    v2f a = *(const v2f*)(xrow + 4 * s);
    acc = __builtin_amdgcn_wmma_f32_16x16x4_f32(false, a, false, breg[s],
                                                (short)0, acc, false, false);
  }

  int nIdx = colBase + mlane;
  int mBase = (lane >= 16) ? 8 : 0;
#pragma unroll
  for (int j = 0; j < 8; ++j)
    xw1[(size_t)(row0 + mBase + j) * H1C + nIdx] = acc[j];
}

// ---------------- GCN layer-1 self term + bias ----------------
__global__ void k_self1(const float* __restrict__ xw1, const float* __restrict__ dis,
                        const float* __restrict__ b1, float* __restrict__ agg1, int n) {
  int i = blockIdx.x * blockDim.x + threadIdx.x;
  if (i >= n * H1C) return;
  int node = i >> 6, f = i & 63;
  float d = dis[node];
  agg1[i] = xw1[i] * d * d + b1[f];
}

// ---------------- GCN layer-1 edge aggregation (atomic, float4 per 16 lanes/edge) ----
__global__ void k_edge1(const int* __restrict__ row, const int* __restrict__ col,
                        const float* __restrict__ dis, const float* __restrict__ xw1,
                        float* __restrict__ agg1, int E) {
  int t = blockIdx.x * blockDim.x + threadIdx.x;
  int e = t >> 4;
  if (e >= E) return;
  int r = row[e], c = col[e];
  if (r == c) return;
  int f4 = (t & 15) * 4;
  float coef = dis[r] * dis[c];
  float4 v = *(const float4*)(xw1 + (size_t)r * H1C + f4);
  float* dst = agg1 + (size_t)c * H1C + f4;
  atomicAdd(dst + 0, v.x * coef);
  atomicAdd(dst + 1, v.y * coef);
  atomicAdd(dst + 2, v.z * coef);
  atomicAdd(dst + 3, v.w * coef);
}

// ---------------- layer-2 GEMV: xw2 = x1 @ W2 (64 -> 1), wave per node ----------------
__global__ void k_gemv2(const float* __restrict__ x1, const float* __restrict__ W2,
                        float* __restrict__ xw2, int n) {
  int waveId = (blockIdx.x * blockDim.x + threadIdx.x) >> 5;
  int lane = threadIdx.x & 31;
  if (waveId >= n) return;
  const float* rp = x1 + (size_t)waveId * H1C;
  float v = rp[lane] * W2[lane] + rp[lane + 32] * W2[lane + 32];
#pragma unroll
  for (int off = 16; off > 0; off >>= 1) v += __shfl_down(v, off, 32);
  if (lane == 0) xw2[waveId] = v;
}

__global__ void k_self2(const float* __restrict__ xw2, const float* __restrict__ dis,
                        const float* __restrict__ b2, float* __restrict__ agg2, int n) {
  int i = blockIdx.x * blockDim.x + threadIdx.x;
  if (i >= n) return;
  float d = dis[i];
  agg2[i] = xw2[i] * d * d + b2[0];
}

__global__ void k_edge2(const int* __restrict__ row, const int* __restrict__ col,
                        const float* __restrict__ dis, const float* __restrict__ xw2,
                        float* __restrict__ agg2, int E) {
  int e = blockIdx.x * blockDim.x + threadIdx.x;
  if (e >= E) return;
  int r = row[e], c = col[e];
  if (r == c) return;
  atomicAdd(&agg2[c], xw2[r] * dis[r] * dis[c]);
}

// ---------------- per-graph counts / starts ----------------
__global__ void k_counts(const int* __restrict__ batch, int* __restrict__ counts, int n) {
  int i = blockIdx.x * blockDim.x + threadIdx.x;
  if (i < n) atomicAdd(&counts[batch[i]], 1);
}
__global__ void k_scan(const int* __restrict__ counts, int* __restrict__ starts, int nb) {
  if (blockIdx.x == 0 && threadIdx.x == 0) {
    int acc = 0;
    for (int g = 0; g < nb; ++g) { starts[g] = acc; acc += counts[g]; }
  }
}

// ---------------- SortPool: top-K by (score desc, index asc), block per graph ----------
__global__ void k_pool(const float* __restrict__ x1, const float* __restrict__ x2,
                       const int* __restrict__ starts, const int* __restrict__ counts,
                       float* __restrict__ p) {
  int g = blockIdx.x;
  int start = starts[g], cnt = counts[g];
  __shared__ float sS[256];
  __shared__ int sI[256];
  __shared__ float prevS;
  __shared__ int prevI;
  if (threadIdx.x == 0) { prevS = INFINITY; prevI = -1; }
  __syncthreads();
  int kmax = cnt < KTOP ? cnt : KTOP;
  for (int j = 0; j < kmax; ++j) {
    float bestS = -INFINITY;
    int bestI = -1;
    float ps = prevS; int pi = prevI;
    for (int t = threadIdx.x; t < cnt; t += blockDim.x) {
      int node = start + t;
      float s = x2[node];
      bool elig = (s < ps) || (s == ps && node > pi);
      if (elig && (s > bestS || (s == bestS && node < bestI))) { bestS = s; bestI = node; }
    }
    sS[threadIdx.x] = bestS; sI[threadIdx.x] = bestI;
    __syncthreads();
    for (int off = 128; off > 0; off >>= 1) {
      if (threadIdx.x < off) {
        float s2 = sS[threadIdx.x + off]; int i2 = sI[threadIdx.x + off];
        float s1 = sS[threadIdx.x];       int i1 = sI[threadIdx.x];
        if (i2 != -1 && (i1 == -1 || s2 > s1 || (s2 == s1 && i2 < i1))) {
          sS[threadIdx.x] = s2; sI[threadIdx.x] = i2;
        }
      }
      __syncthreads();
    }
    int node = sI[0];
    if (threadIdx.x == 0) { prevS = sS[0]; prevI = node; }
    __syncthreads();
    if (node >= 0) {
      for (int f = threadIdx.x; f < PCOLS; f += blockDim.x)
        p[(size_t)g * PLEN + j * PCOLS + f] =
            (f < H1C) ? x1[(size_t)node * H1C + f] : x2[node];
    }
    __syncthreads();
  }
}

// ---------------- CNN head: conv97/s97 + relu + maxpool2 + conv5 + relu ----------------
__global__ void k_head(const float* __restrict__ p, const float* __restrict__ w3,
                       const float* __restrict__ b3, const float* __restrict__ w4,
                       const float* __restrict__ b4, float* __restrict__ flat) {
  int g = blockIdx.x;
  __shared__ float sp[PLEN];
  __shared__ float sh[16 * 20];
  __shared__ float sm[16 * 10];
  for (int i = threadIdx.x; i < PLEN; i += blockDim.x) sp[i] = p[(size_t)g * PLEN + i];
  __syncthreads();
  for (int idx = threadIdx.x; idx < 16 * 20; idx += blockDim.x) {
    int oc = idx / 20, t = idx % 20;
    float acc = b3[oc];
    const float* w = w3 + oc * 97;
    int base = t * 97;
    for (int j = 0; j < 97; ++j) acc += sp[base + j] * w[j];
    sh[idx] = fmaxf(acc, 0.0f);
  }
  __syncthreads();
  for (int idx = threadIdx.x; idx < 16 * 10; idx += blockDim.x) {
    int oc = idx / 10, t = idx % 10;
    sm[idx] = fmaxf(sh[oc * 20 + 2 * t], sh[oc * 20 + 2 * t + 1]);
  }
  __syncthreads();
  for (int idx = threadIdx.x; idx < 32 * 6; idx += blockDim.x) {
    int oc = idx / 6, t = idx % 6;
    float acc = b4[oc];
    const float* w = w4 + oc * 80;
    for (int ic = 0; ic < 16; ++ic)
#pragma unroll
      for (int j = 0; j < 5; ++j) acc += sm[ic * 10 + t + j] * w[ic * 5 + j];
    flat[(size_t)g * FLAT2 + idx] = fmaxf(acc, 0.0f);
  }
}

// ---------------- WMMA fc1: last = relu(flat @ fc1_w + fc1_b), (512-pad)x192x128 --------
// blockDim = 256 (8 waves); block handles 16 rows; wave w handles col tile w (8 tiles).
// flat is padded/zeroed to MPAD rows so all loads are unconditional.
__global__ void k_fc1(const float* __restrict__ flat, const float* __restrict__ w,
                      const float* __restrict__ b, float* __restrict__ last,
                      float* __restrict__ out_last, int M) {
  int wave = threadIdx.x >> 5;
  int lane = threadIdx.x & 31;
  int row0 = blockIdx.x * 16;
  int colBase = wave * 16;
  int mlane = lane & 15;
  int khalf = (lane >= 16) ? 2 : 0;

  v2f breg[48];
#pragma unroll
  for (int s = 0; s < 48; ++s) {
    int ka = 4 * s + khalf;
    breg[s].x = w[(size_t)ka * 128 + colBase + mlane];
    breg[s].y = w[(size_t)(ka + 1) * 128 + colBase + mlane];
  }

  const float* arow = flat + (size_t)(row0 + mlane) * FLAT2 + khalf;
  v8f acc = {};
#pragma unroll
  for (int s = 0; s < 48; ++s) {
    v2f a = *(const v2f*)(arow + 4 * s);
    acc = __builtin_amdgcn_wmma_f32_16x16x4_f32(false, a, false, breg[s],
                                                (short)0, acc, false, false);
  }

  int nIdx = colBase + mlane;
  float bias = b[nIdx];
  int mBase = (lane >= 16) ? 8 : 0;
#pragma unroll
  for (int j = 0; j < 8; ++j) {
    int rr = row0 + mBase + j;
    float v = fmaxf(acc[j] + bias, 0.0f);
    last[(size_t)rr * 128 + nIdx] = v;           // padded buffer, always safe
    if (rr < M) out_last[(size_t)rr * 128 + nIdx] = v;
  }
}

// ---------------- fc2 + log_softmax; writes classes (off 0) and out (off 9000) ----------
__global__ void k_fc2(const float* __restrict__ last, const float* __restrict__ w,
                      const float* __restrict__ b, float* __restrict__ out) {
  int g = blockIdx.x;
  __shared__ float sv[NCLS];
  int t = threadIdx.x;
  if (t < NCLS) {
    float acc = b[t];
    const float* L = last + (size_t)g * 128;
    for (int i = 0; i < 128; ++i) acc += L[i] * w[i * NCLS + t];
    sv[t] = acc;
  }
  __syncthreads();
  if (t < NCLS) {
    float m = sv[0];
    for (int i = 1; i < NCLS; ++i) m = fmaxf(m, sv[i]);
    float s = 0.0f;
    for (int i = 0; i < NCLS; ++i) s += expf(sv[i] - m);
    float lse = m + logf(s);
    out[(size_t)g * NCLS + t] = sv[t] - lse;          // classes (log_softmax)
    out[9000 + (size_t)g * NCLS + t] = sv[t];          // raw logits
  }
}

extern "C" void kernel_launch(void* const* d_in, const int* in_sizes, int n_in,
                              void* d_out, int out_size, void* d_ws, size_t ws_size,
                              hipStream_t stream) {
  const float* x   = (const float*)d_in[0];
  const int*   ei  = (const int*)d_in[1];
  const int*   bat = (const int*)d_in[2];
  const float* W1  = (const float*)d_in[4];
  const float* b1  = (const float*)d_in[5];
  const float* W2  = (const float*)d_in[6];
  const float* b2  = (const float*)d_in[7];
  const float* w3  = (const float*)d_in[8];
  const float* b3  = (const float*)d_in[9];
  const float* w4  = (const float*)d_in[10];
  const float* b4  = (const float*)d_in[11];
  const float* fw1 = (const float*)d_in[12];
  const float* fb1 = (const float*)d_in[13];
  const float* fw2 = (const float*)d_in[14];
  const float* fb2 = (const float*)d_in[15];
  float* out = (float*)d_out;

  const int N = in_sizes[0] / F_IN;   // 100000, multiple of 16
  const int E = in_sizes[1] / 2;
  const int B = NGR;
  const int* row = ei;
  const int* col = ei + E;

  float* ws   = (float*)d_ws;
  float* xw1  = ws;
  float* agg1 = xw1 + (size_t)N * H1C;     // becomes x1 after tanh
  float* deg  = agg1 + (size_t)N * H1C;
  float* dis  = deg + N;
  float* xw2  = dis + N;
  float* agg2 = xw2 + N;                   // becomes x2 after tanh
  float* p    = agg2 + N;
  float* flat = p + (size_t)B * PLEN;      // MPAD x 192, zero-padded
  float* last = flat + (size_t)MPAD * FLAT2;  // MPAD x 128
  int* counts = (int*)(last + (size_t)MPAD * 128);
  int* starts = counts + B;

  auto cdiv = [](long long a, long long b) { return (unsigned)((a + b - 1) / b); };

  k_fill_f<<<cdiv(N, 256), 256, 0, stream>>>(deg, N, 1.0f);
  k_fill_f<<<cdiv((long long)B * PLEN, 256), 256, 0, stream>>>(p, B * PLEN, 0.0f);
  k_fill_f<<<cdiv((long long)MPAD * FLAT2, 256), 256, 0, stream>>>(flat, MPAD * FLAT2, 0.0f);
  k_zero_i<<<cdiv(B, 256), 256, 0, stream>>>(counts, B);

  k_deg<<<cdiv(E, 256), 256, 0, stream>>>(row, col, deg, E);
  k_dis<<<cdiv(N, 256), 256, 0, stream>>>(deg, dis, N);

  k_gemm1<<<cdiv(N, 16), 128, 0, stream>>>(x, W1, xw1);
  k_self1<<<cdiv((long long)N * H1C, 256), 256, 0, stream>>>(xw1, dis, b1, agg1, N);
  k_edge1<<<cdiv((long long)E * 16, 256), 256, 0, stream>>>(row, col, dis, xw1, agg1, E);
  k_tanh<<<cdiv((long long)N * H1C, 256), 256, 0, stream>>>(agg1, N * H1C);

  k_gemv2<<<cdiv((long long)N * 32, 256), 256, 0, stream>>>(agg1, W2, xw2, N);
  k_self2<<<cdiv(N, 256), 256, 0, stream>>>(xw2, dis, b2, agg2, N);
  k_edge2<<<cdiv(E, 256), 256, 0, stream>>>(row, col, dis, xw2, agg2, E);
  k_tanh<<<cdiv(N, 256), 256, 0, stream>>>(agg2, N);

  k_counts<<<cdiv(N, 256), 256, 0, stream>>>(bat, counts, N);
  k_scan<<<1, 1, 0, stream>>>(counts, starts, B);
  k_pool<<<B, 256, 0, stream>>>(agg1, agg2, starts, counts, p);

  k_head<<<B, 256, 0, stream>>>(p, w3, b3, w4, b4, flat);
  k_fc1<<<MPAD / 16, 256, 0, stream>>>(flat, fw1, fb1, last, out + 18000, B);
  k_fc2<<<B, 32, 0, stream>>>(last, fw2, fb2, out);
}